// Encoder_28269474742836
// MI455X (gfx1250) — compile-verified
//
#include <hip/hip_runtime.h>

#define DD 128
#define BN_EPS 1e-5f

typedef float v2f __attribute__((ext_vector_type(2)));
typedef float v8f __attribute__((ext_vector_type(8)));

// ---------------- degree / norm ----------------
__global__ __launch_bounds__(256) void k_init(float* __restrict__ deg,
                                              float* __restrict__ stats, int n) {
  int i = blockIdx.x * blockDim.x + threadIdx.x;
  if (i < n) deg[i] = 1.0f;          // self-loop contribution
  if (i < 512) stats[i] = 0.0f;      // zero all stat accumulators
}

__global__ __launch_bounds__(256) void k_deg(const long long* __restrict__ dst,
                                             float* __restrict__ deg, int nE) {
  int i = blockIdx.x * blockDim.x + threadIdx.x;
  if (i < nE) unsafeAtomicAdd(&deg[(int)dst[i]], 1.0f);
}

__global__ __launch_bounds__(256) void k_dinv(float* __restrict__ deg, int n) {
  int i = blockIdx.x * blockDim.x + threadIdx.x;
  if (i < n) {
    float d = deg[i];
    deg[i] = d > 0.0f ? rsqrtf(d) : 0.0f;
  }
}

// ---------------- GEMM: H = X @ W  (FP32 WMMA 16x16x4) ----------------
// Block = 256 threads = 8 waves; block owns a 16-row tile, wave w owns cols [16w,16w+16).
__global__ __launch_bounds__(256) void k_gemm(const float* __restrict__ X,
                                              const float* __restrict__ W,
                                              float* __restrict__ H) {
  __shared__ float xs[16 * 129];               // padded stride vs bank conflicts
  const int lane = threadIdx.x & 31;
  const int wave = threadIdx.x >> 5;
  const int rowbase = blockIdx.x << 4;

  for (int i = threadIdx.x; i < 16 * DD; i += 256) {
    int r = i >> 7, c = i & 127;
    xs[r * 129 + c] = X[(size_t)(rowbase + r) * DD + c];
  }
  __syncthreads();

  const int colbase = wave << 4;
  const int l15 = lane & 15;
  const int koff = (lane >> 4) << 1;           // lanes 0-15 -> K{0,1}, 16-31 -> K{2,3}
  const float* xrow = &xs[l15 * 129];
  v8f acc = {0.f, 0.f, 0.f, 0.f, 0.f, 0.f, 0.f, 0.f};

#pragma unroll 8
  for (int k0 = 0; k0 < DD; k0 += 4) {
    int k = k0 + koff;
    v2f a, b;
    a.x = xrow[k];
    a.y = xrow[k + 1];
    b.x = W[(size_t)k * DD + colbase + l15];
    b.y = W[(size_t)(k + 1) * DD + colbase + l15];
    acc = __builtin_amdgcn_wmma_f32_16x16x4_f32(false, a, false, b, (short)0, acc,
                                                false, false);
  }

  const int rowoff = (lane >> 4) << 3;         // VGPR i holds rows i / i+8
#pragma unroll
  for (int i = 0; i < 8; ++i)
    H[(size_t)(rowbase + rowoff + i) * DD + colbase + l15] = acc[i];
}

// ---------------- AGG init with self-loop term ----------------
__global__ __launch_bounds__(256) void k_selfloop(const float* __restrict__ H,
                                                  const float* __restrict__ dinv,
                                                  float* __restrict__ AGG, int total) {
  int i = blockIdx.x * blockDim.x + threadIdx.x;
  if (i < total) {
    float di = dinv[i >> 7];
    AGG[i] = di * di * H[i];
  }
}

// ---------------- edge scatter: one wave per edge, float4 per lane ----------------
__global__ __launch_bounds__(256) void k_scatter(const long long* __restrict__ src,
                                                 const long long* __restrict__ dst,
                                                 const float* __restrict__ dinv,
                                                 const float* __restrict__ H,
                                                 float* __restrict__ AGG, int nE) {
  const int lane = threadIdx.x & 31;
  int w = (blockIdx.x * blockDim.x + threadIdx.x) >> 5;
  int nw = (gridDim.x * blockDim.x) >> 5;
  for (int e = w; e < nE; e += nw) {
    if (e + nw < nE) {
      __builtin_prefetch(&src[e + nw], 0, 0);
      __builtin_prefetch(&dst[e + nw], 0, 0);
    }
    int s = (int)src[e];
    int d = (int)dst[e];
    float norm = dinv[s] * dinv[d];
    float4 v = ((const float4*)(H + (size_t)s * DD))[lane];
    float* out = AGG + (size_t)d * DD + (lane << 2);
    unsafeAtomicAdd(out + 0, norm * v.x);
    unsafeAtomicAdd(out + 1, norm * v.y);
    unsafeAtomicAdd(out + 2, norm * v.z);
    unsafeAtomicAdd(out + 3, norm * v.w);
  }
}

// ---------------- bias add + column sum/sumsq (block-local reduce) ----------------
__global__ __launch_bounds__(128) void k_bias_stats(float* __restrict__ AGG,
                                                    const float* __restrict__ bias,
                                                    float* __restrict__ sum,
                                                    float* __restrict__ sq,
                                                    int rows_per_blk, int n) {
  int c = threadIdx.x;                       // column 0..127
  int r0 = blockIdx.x * rows_per_blk;
  int r1 = r0 + rows_per_blk; if (r1 > n) r1 = n;
  float b = bias[c];
  float s = 0.f, q = 0.f;
  for (int r = r0; r < r1; ++r) {
    float v = AGG[(size_t)r * DD + c] + b;
    AGG[(size_t)r * DD + c] = v;
    s += v;
    q += v * v;
  }
  unsafeAtomicAdd(&sum[c], s);
  unsafeAtomicAdd(&sq[c], q);
}

// ---------------- BatchNorm (batch stats) + PReLU ----------------
__global__ __launch_bounds__(256) void k_bn_prelu(const float* __restrict__ AGG,
                                                  const float* __restrict__ sum,
                                                  const float* __restrict__ sq,
                                                  const float* __restrict__ gamma,
                                                  const float* __restrict__ beta,
                                                  const float* __restrict__ alpha,
                                                  float* __restrict__ OUT,
                                                  int total, float invN) {
  int i = blockIdx.x * blockDim.x + threadIdx.x;
  if (i < total) {
    int c = i & 127;
    float mu = sum[c] * invN;
    float var = sq[c] * invN - mu * mu;
    float y = (AGG[i] - mu) * rsqrtf(var + BN_EPS) * gamma[c] + beta[c];
    float a = alpha[0];
    OUT[i] = y > 0.f ? y : a * y;
  }
}

extern "C" void kernel_launch(void* const* d_in, const int* in_sizes, int n_in,
                              void* d_out, int out_size, void* d_ws, size_t ws_size,
                              hipStream_t stream) {
  const float* x        = (const float*)d_in[0];
  const long long* ei   = (const long long*)d_in[1];   // int64 [2, E]
  const float* W1  = (const float*)d_in[2];
  const float* b1  = (const float*)d_in[3];
  const float* g1  = (const float*)d_in[4];
  const float* be1 = (const float*)d_in[5];
  const float* a1  = (const float*)d_in[6];
  const float* W2  = (const float*)d_in[7];
  const float* b2  = (const float*)d_in[8];
  const float* g2  = (const float*)d_in[9];
  const float* be2 = (const float*)d_in[10];
  const float* a2  = (const float*)d_in[11];

  const int n  = in_sizes[0] / DD;   // 50000
  const int nE = in_sizes[1] / 2;    // 800000
  const long long* srcE = ei;
  const long long* dstE = ei + nE;

  char* ws   = (char*)d_ws;
  float* H   = (float*)ws;                                 // n*128 floats
  float* AGG = (float*)(ws + (size_t)n * DD * 4);          // n*128 floats
  float* DEG = (float*)(ws + (size_t)2 * n * DD * 4);      // n floats (-> dinv)
  float* ST  = DEG + n;                                    // 512 floats of stats
  float* sum1 = ST, * sq1 = ST + 128, * sum2 = ST + 256, * sq2 = ST + 384;
  float* OUT = (float*)d_out;

  const int total = n * DD;
  const float invN = 1.0f / (float)n;
  const int rpb = 128;

  // degrees -> dinv
  k_init<<<(n + 255) / 256, 256, 0, stream>>>(DEG, ST, n);
  k_deg<<<(nE + 255) / 256, 256, 0, stream>>>(dstE, DEG, nE);
  k_dinv<<<(n + 255) / 256, 256, 0, stream>>>(DEG, n);

  // ---- layer 1 ----
  k_gemm<<<n / 16, 256, 0, stream>>>(x, W1, H);
  k_selfloop<<<(total + 255) / 256, 256, 0, stream>>>(H, DEG, AGG, total);
  k_scatter<<<4096, 256, 0, stream>>>(srcE, dstE, DEG, H, AGG, nE);
  k_bias_stats<<<(n + rpb - 1) / rpb, 128, 0, stream>>>(AGG, b1, sum1, sq1, rpb, n);
  k_bn_prelu<<<(total + 255) / 256, 256, 0, stream>>>(AGG, sum1, sq1, g1, be1, a1,
                                                      OUT, total, invN);

  // ---- layer 2 (input is layer-1 output living in d_out) ----
  k_gemm<<<n / 16, 256, 0, stream>>>(OUT, W2, H);
  k_selfloop<<<(total + 255) / 256, 256, 0, stream>>>(H, DEG, AGG, total);
  k_scatter<<<4096, 256, 0, stream>>>(srcE, dstE, DEG, H, AGG, nE);
  k_bias_stats<<<(n + rpb - 1) / rpb, 128, 0, stream>>>(AGG, b2, sum2, sq2, rpb, n);
  k_bn_prelu<<<(total + 255) / 256, 256, 0, stream>>>(AGG, sum2, sq2, g2, be2, a2,
                                                      OUT, total, invN);
}